// OpMessagePassing_42666205119386
// MI455X (gfx1250) — compile-verified
//
#include <hip/hip_runtime.h>

typedef __attribute__((ext_vector_type(2))) float v2f;
typedef __attribute__((ext_vector_type(8))) float v8f;

#define D_FEAT 64
#define WAVES_PER_BLOCK 8

// One wave32 computes a 16-row x 64-col output tile as S(16xK) x G(Kx64)
// using V_WMMA_F32_16X16X4_F32, K chunked by 4. acd0 is sorted, so the
// events for rows [T0, T0+16) are a contiguous range [mlo, mhi).
// All per-chunk descriptor traffic is wave-uniform (SGPR/SMEM); only the
// Aval row gathers and the WMMA operands live in VGPRs. No atomics; every
// output row is written exactly once, non-temporally.
__global__ __launch_bounds__(256) void spspmm_wmma_kernel(
    const float* __restrict__ Aval,   // [NNZ_A, 64]
    const float* __restrict__ Bval,   // [NNZ_B]
    const int*   __restrict__ acd0,   // [M] sorted output indices
    const int*   __restrict__ acd1,   // [M] indices into Aval rows
    const int*   __restrict__ acd2,   // [M] indices into Bval
    float*       __restrict__ out,    // [tar_nnz, 64]
    int M, int tar_nnz)
{
    const int lane = threadIdx.x & 31;
    const int tile = blockIdx.x * WAVES_PER_BLOCK + (threadIdx.x >> 5);
    const int T0   = tile * 16;
    if (T0 >= tar_nnz) return;   // uniform per wave -> EXEC stays all-1s

    // Dual binary search: lanes 0-15 find lower_bound(T0),
    // lanes 16-31 find lower_bound(T0+16). One pass instead of two.
    const int key = T0 + (lane & 16);
    int lo = 0, hi = M;
    while (lo < hi) {
        const int mid = (lo + hi) >> 1;
        const bool lt = (acd0[mid] < key);
        lo = lt ? (mid + 1) : lo;
        hi = lt ? hi : mid;
    }
    // Pin the bounds into SGPRs: everything derived from them is uniform.
    const int mlo = __builtin_amdgcn_readlane(lo, 0);
    const int mhi = __builtin_amdgcn_readlane(lo, 16);

    v8f c0 = {}, c1 = {}, c2 = {}, c3 = {};

    const int row = lane & 15;          // A-matrix M index for this lane
    const int kk  = (lane >> 4) << 1;   // K pair selected by lane half: 0 or 2
    const int n   = lane & 15;          // column within a 16-wide slab

    // Chunks aligned to 4 events -> aligned b128 scalar index loads.
    // Partial events at either edge are killed by coef = 0 (scalar select);
    // their seg/arow values are harmless (coef 0 annihilates the product).
    const int mbase = mlo & ~3;
    for (int mb = mbase; mb < mhi; mb += 4) {
        const int4 s0 = *(const int4*)(acd0 + mb);   // uniform -> s_load_b128
        const int4 s1 = *(const int4*)(acd1 + mb);
        const int4 s2 = *(const int4*)(acd2 + mb);

        const int seg[4]  = { s0.x, s0.y, s0.z, s0.w };
        const int arow[4] = { s1.x, s1.y, s1.z, s1.w };
        const int bidx[4] = { s2.x, s2.y, s2.z, s2.w };

        float coef[4];
#pragma unroll
        for (int k = 0; k < 4; ++k) {
            const int mk = mb + k;                       // uniform
            const bool ok = (mk >= mlo) & (mk < mhi);    // scalar bool
            const float bv = Bval[bidx[k]];              // uniform s_load
            coef[k] = ok ? bv : 0.0f;                    // s_cselect
        }

        // A fragment (16x4 f32): A[t,k] = coef[k] * (acd0[k] == T0+t)
        v2f a;
        a.x = (seg[kk]     == T0 + row) ? coef[kk]     : 0.0f;
        a.y = (seg[kk + 1] == T0 + row) ? coef[kk + 1] : 0.0f;

        // B fragments (4x16 f32 each): gathered Aval rows, 4 channel slabs.
        // arow[] is an SGPR -> saddr-form global_load_b32 with lane offset n.
        const float* __restrict__ r0 = Aval + (size_t)(unsigned)arow[kk]     * D_FEAT;
        const float* __restrict__ r1 = Aval + (size_t)(unsigned)arow[kk + 1] * D_FEAT;
        v2f b0, b1, b2, b3;
        b0.x = r0[n];      b0.y = r1[n];
        b1.x = r0[16 + n]; b1.y = r1[16 + n];
        b2.x = r0[32 + n]; b2.y = r1[32 + n];
        b3.x = r0[48 + n]; b3.y = r1[48 + n];

        c0 = __builtin_amdgcn_wmma_f32_16x16x4_f32(false, a, false, b0, (short)0, c0, false, false);
        c1 = __builtin_amdgcn_wmma_f32_16x16x4_f32(false, a, false, b1, (short)0, c1, false, false);
        c2 = __builtin_amdgcn_wmma_f32_16x16x4_f32(false, a, false, b2, (short)0, c2, false, false);
        c3 = __builtin_amdgcn_wmma_f32_16x16x4_f32(false, a, false, b3, (short)0, c3, false, false);
    }

    // Store C/D tile: VGPR j holds row j (lanes 0-15) or row j+8 (lanes 16-31).
    // One base address per lane; all 32 stores use immediate offsets.
    // Non-temporal: output is write-once, keep L2 for the Aval gathers.
    const unsigned rbase = (unsigned)T0 + ((unsigned)lane >> 4) * 8u;
    float* o = out + (size_t)rbase * D_FEAT + (unsigned)n;
    if (T0 + 16 <= tar_nnz) {          // uniform fast path: full tile
#pragma unroll
        for (int j = 0; j < 8; ++j) {
            __builtin_nontemporal_store(c0[j], o + j * D_FEAT + 0);
            __builtin_nontemporal_store(c1[j], o + j * D_FEAT + 16);
            __builtin_nontemporal_store(c2[j], o + j * D_FEAT + 32);
            __builtin_nontemporal_store(c3[j], o + j * D_FEAT + 48);
        }
    } else {                           // partial last tile (not hit for 1M rows)
#pragma unroll
        for (int j = 0; j < 8; ++j) {
            if ((int)(rbase + j) < tar_nnz) {
                __builtin_nontemporal_store(c0[j], o + j * D_FEAT + 0);
                __builtin_nontemporal_store(c1[j], o + j * D_FEAT + 16);
                __builtin_nontemporal_store(c2[j], o + j * D_FEAT + 32);
                __builtin_nontemporal_store(c3[j], o + j * D_FEAT + 48);
            }
        }
    }
}

extern "C" void kernel_launch(void* const* d_in, const int* in_sizes, int n_in,
                              void* d_out, int out_size, void* d_ws, size_t ws_size,
                              hipStream_t stream) {
    const float* Aval = (const float*)d_in[0];
    const float* Bval = (const float*)d_in[1];
    const int*   acd0 = (const int*)d_in[2];
    const int*   acd1 = (const int*)d_in[3];
    const int*   acd2 = (const int*)d_in[4];
    float*       out  = (float*)d_out;

    const int M       = in_sizes[2];          // number of multiplication events
    const int tar_nnz = out_size / D_FEAT;    // output rows

    const int tiles  = (tar_nnz + 15) / 16;
    const int blocks = (tiles + WAVES_PER_BLOCK - 1) / WAVES_PER_BLOCK;
    spspmm_wmma_kernel<<<blocks, WAVES_PER_BLOCK * 32, 0, stream>>>(
        Aval, Bval, acd0, acd1, acd2, out, M, tar_nnz);
}